// SMC_Transf_Cell_80564996539074
// MI455X (gfx1250) — compile-verified
//
#include <hip/hip_runtime.h>
#include <math.h>

// Problem constants (from reference)
#define BATCH 32
#define PART  10
#define BP    320          // BATCH*PART
#define DM    256
#define DFF_  1024
#define OUTD  32
#define SEQL  512
#define CH    16           // seq rows per TDM chunk
#define NCH   (SEQL / CH)  // 32 chunks

typedef __attribute__((ext_vector_type(2))) float v2f;
typedef __attribute__((ext_vector_type(4))) float f4;
typedef __attribute__((ext_vector_type(8))) float v8f;
typedef __attribute__((ext_vector_type(4))) unsigned int v4u;
typedef __attribute__((ext_vector_type(4))) int v4i;
typedef __attribute__((ext_vector_type(8))) int v8i;

__device__ __forceinline__ float wave_sum(float v) {
#pragma unroll
  for (int off = 16; off >= 1; off >>= 1) v += __shfl_xor(v, off, 32);
  return v;
}

// ---------------------------------------------------------------------------
// TDM: issue a 2D tile load (rows x DM f32, row-major, contiguous) into LDS.
// D# packing per CDNA5 ISA (groups 0..3).
// Caller must be wave-uniform; completion tracked by TENSORcnt.
// ---------------------------------------------------------------------------
__device__ __forceinline__ void tdm_load_rows(const float* gsrc,
                                              unsigned lds_addr, int rows)
{
  const unsigned long long ga = (unsigned long long)(uintptr_t)gsrc;
  v4u g0;
  g0.x = 1u;                                        // count=1 (valid user D#)
  g0.y = lds_addr;                                  // LDS byte address
  g0.z = (unsigned)ga;                              // global_addr[31:0]
  g0.w = (unsigned)((ga >> 32) & 0x1FFFFFFu)        // global_addr[56:32]
         | (2u << 30);                              // type=2 ("image")
  v8i g1;
  g1[0] = (int)(2u << 16);                          // data_size=2 -> 4 bytes
  g1[1] = (int)(((unsigned)DM & 0xFFFFu) << 16);    // tensor_dim0[15:0]
  g1[2] = (int)(((unsigned)SEQL & 0xFFFFu) << 16);  // dim0_hi=0 | tensor_dim1[15:0]
  g1[3] = (int)((unsigned)DM << 16);                // dim1_hi=0 | tile_dim0=DM
  g1[4] = (int)((unsigned)rows & 0xFFFFu);          // tile_dim1=rows, tile_dim2=0
  g1[5] = (int)DM;                                  // tensor_dim0_stride low32
  g1[6] = 0;                                        // stride0_hi | stride1_lo
  g1[7] = 0;
  v4i g2 = {0, 0, 0, 0};                            // 2D: groups 2/3 unused
  v4i g3 = {0, 0, 0, 0};
  v8i g4 = {0, 0, 0, 0, 0, 0, 0, 0};                // unused trailing group (clang-23 6-arg form)
  __builtin_amdgcn_tensor_load_to_lds(g0, g1, g2, g3, g4, 0);
}

// ---------------------------------------------------------------------------
// Generic M-tile-16 GEMM on the WMMA f32 pipe:
//   C[m0:m0+16, :] = epilogue( A[m0:m0+16, :K] @ W[K x N] + bias )
// epilogue: (+ noise_scale*noise) -> relu? -> (+residual, LayerNorm)?
// Block: 256 threads = 8 waves; wave w owns N-tiles w, w+8, ...
// STAGEA templated so A-fragment loads are pure DS or pure GLOBAL (no FLAT).
// ---------------------------------------------------------------------------
template<bool STAGEA>
__global__ __launch_bounds__(256)
void gemm16_wmma(const float* __restrict__ A, const float* __restrict__ W,
                 const float* __restrict__ bias,
                 const float* __restrict__ noise, float noise_scale,
                 int do_relu,
                 const float* __restrict__ residual,    // non-null => fused LN
                 const float* __restrict__ ln_g, const float* __restrict__ ln_b,
                 float* __restrict__ C, int N, int K)
{
  extern __shared__ float smem[];
  const int tid  = threadIdx.x;
  const int lane = tid & 31;
  const int wave = tid >> 5;
  const int l15  = lane & 15;
  const int half = lane >> 4;          // 0: K-pair {0,1}, 1: K-pair {2,3}
  const int m0   = blockIdx.x << 4;
  const int LDA  = K + 2;              // pad to spread LDS banks
  const int csbase = STAGEA ? 16 * LDA : 0;
  const bool hasLN = (residual != nullptr);

  if (STAGEA) {
    for (int idx = tid; idx < 16 * K; idx += 256) {
      int r = idx / K, c = idx - r * K;
      smem[r * LDA + c] = A[(size_t)(m0 + r) * K + c];
    }
  }
  __syncthreads();

  const float* Ag = A + (size_t)(m0 + l15) * K;   // used when !STAGEA
  const int arow = l15 * LDA;                     // used when STAGEA
  const int nTiles = N >> 4;
  for (int nt = wave; nt < nTiles; nt += 8) {
    const int n0 = nt << 4;
    v8f acc = {0.f, 0.f, 0.f, 0.f, 0.f, 0.f, 0.f, 0.f};
    for (int kk = 0; kk < K; kk += 4) {
      const int ka = kk + (half << 1);
      v2f a, b;
      if (STAGEA) { a.x = smem[arow + ka]; a.y = smem[arow + ka + 1]; }
      else        { a.x = Ag[ka];          a.y = Ag[ka + 1]; }
      const float* wp = W + (size_t)ka * N + n0 + l15;
      b.x = wp[0];
      b.y = wp[N];
      // D = A(16x4) * B(4x16) + C  (exact f32)
      acc = __builtin_amdgcn_wmma_f32_16x16x4_f32(false, a, false, b,
                                                  (short)0, acc, false, false);
    }
    const int col = n0 + l15;
    const float bi = bias ? bias[col] : 0.f;
#pragma unroll
    for (int vg = 0; vg < 8; ++vg) {
      const int row = vg + (half << 3);
      const size_t gidx = (size_t)(m0 + row) * N + col;
      float v = acc[vg] + bi;
      if (noise) v += noise_scale * noise[gidx];
      if (do_relu) v = fmaxf(v, 0.f);
      if (hasLN) {
        smem[csbase + row * N + col] = v + residual[gidx];
      } else {
        C[gidx] = v;
      }
    }
  }

  if (hasLN) {
    __syncthreads();
    __shared__ float ps[256], ps2[256], mu_s[16], rs_s[16];
    const int r  = tid >> 4;           // 16 threads cooperate per row
    const int c0 = tid & 15;
    float s = 0.f, s2 = 0.f;
    for (int c = c0; c < N; c += 16) {
      float v = smem[csbase + r * N + c];
      s += v; s2 += v * v;
    }
    ps[tid] = s; ps2[tid] = s2;
    __syncthreads();
    if (tid < 16) {
      float ss = 0.f, ss2 = 0.f;
#pragma unroll
      for (int i = 0; i < 16; ++i) { ss += ps[tid * 16 + i]; ss2 += ps2[tid * 16 + i]; }
      const float mu = ss / (float)N;
      const float var = ss2 / (float)N - mu * mu;
      mu_s[tid] = mu;
      rs_s[tid] = rsqrtf(var + 1e-6f);
    }
    __syncthreads();
    for (int idx = tid; idx < 16 * N; idx += 256) {
      const int rr = idx / N, cc = idx - rr * N;
      C[(size_t)(m0 + rr) * N + cc] =
          (smem[csbase + idx] - mu_s[rr]) * rs_s[rr] * ln_g[cc] + ln_b[cc];
    }
  }
}

// ---------------------------------------------------------------------------
// Per-particle single-query attention, K/V staged through LDS by the TDM.
// 1 block (8 waves) per (b,p). Wave 0 drives the DMA pipeline (double
// buffered CH-row chunks, TENSORcnt + barrier sync); all waves compute.
// K/V row t substituted from the freshly projected k/v (uniform branch).
// ---------------------------------------------------------------------------
__global__ __launch_bounds__(256)
void attn_kernel(const float* __restrict__ Kin, const float* __restrict__ Vin,
                 const float* __restrict__ qw, const float* __restrict__ kw,
                 const float* __restrict__ vw, const int* __restrict__ tptr,
                 float* __restrict__ attn_out, float* __restrict__ z_out)
{
  __shared__ float sbuf[2][CH * DM];     // TDM staging (2 x 16 KB)
  __shared__ float qs[DM];
  __shared__ float lw[SEQL];
  __shared__ float red[256];
  const int bp   = blockIdx.x;
  const int tid  = threadIdx.x;
  const int lane = tid & 31;
  const int wave = tid >> 5;
  const int t    = *tptr;

  qs[tid] = qw[(size_t)bp * DM + tid];

  const float* Kb = Kin + (size_t)bp * SEQL * DM;
  const float* Vb = Vin + (size_t)bp * SEQL * DM;
  const float* kr = kw + (size_t)bp * DM;
  const float* vr = vw + (size_t)bp * DM;
  const unsigned lds0 = (unsigned)(uintptr_t)&sbuf[0][0];
  const unsigned lds1 = (unsigned)(uintptr_t)&sbuf[1][0];

  if (wave == 0) tdm_load_rows(Kb, lds0, CH);      // prime K chunk 0
  __syncthreads();                                 // qs visible

  const f4 q0 = *(const f4*)(qs + 4 * lane);
  const f4 q1 = *(const f4*)(qs + 128 + 4 * lane);

  // ---- phase 1: logits (wave-cooperative dots out of LDS) ----
  for (int c = 0; c < NCH; ++c) {
    if (wave == 0) __builtin_amdgcn_s_wait_tensorcnt(0);
    __syncthreads();                               // chunk c resident
    if (wave == 0 && c + 1 < NCH)
      tdm_load_rows(Kb + (size_t)(c + 1) * CH * DM, (c & 1) ? lds0 : lds1, CH);
    const int bsel = c & 1;
#pragma unroll
    for (int rr = 0; rr < CH / 8; ++rr) {
      const int r = (CH / 8) * wave + rr;
      const int s = c * CH + r;
      f4 x0, x1;
      if (s == t) {            // fresh k row from global (uniform per wave)
        x0 = *(const f4*)(kr + 4 * lane);
        x1 = *(const f4*)(kr + 128 + 4 * lane);
      } else {
        x0 = *(const f4*)&sbuf[bsel][r * DM + 4 * lane];
        x1 = *(const f4*)&sbuf[bsel][r * DM + 128 + 4 * lane];
      }
      float acc = x0.x * q0.x + x0.y * q0.y + x0.z * q0.z + x0.w * q0.w +
                  x1.x * q1.x + x1.y * q1.y + x1.z * q1.z + x1.w * q1.w;
      acc = wave_sum(acc);
      if (lane == 0) lw[s] = acc * 0.0625f;        // 1/sqrt(256)
    }
    __syncthreads();                               // done reading buffer c&1
  }

  if (wave == 0) tdm_load_rows(Vb, lds0, CH);      // prime V chunk 0 (overlaps softmax)

  // ---- softmax over 512 (two elements per thread) ----
  const float l0 = lw[tid], l1 = lw[tid + 256];
  red[tid] = fmaxf(l0, l1);
  __syncthreads();
  for (int off = 128; off > 0; off >>= 1) {
    if (tid < off) red[tid] = fmaxf(red[tid], red[tid + off]);
    __syncthreads();
  }
  const float mx = red[0];
  __syncthreads();
  float e0 = expf(l0 - mx), e1 = expf(l1 - mx);
  red[tid] = e0 + e1;
  __syncthreads();
  for (int off = 128; off > 0; off >>= 1) {
    if (tid < off) red[tid] += red[tid + off];
    __syncthreads();
  }
  const float inv = 1.f / red[0];
  __syncthreads();
  e0 *= inv; e1 *= inv;
  lw[tid] = e0; lw[tid + 256] = e1;
  attn_out[(size_t)bp * SEQL + tid] = e0;
  attn_out[(size_t)bp * SEQL + tid + 256] = e1;
  __syncthreads();

  // ---- phase 2: z = attn @ V (thread owns one model dim) ----
  float acc = 0.f;
  for (int c = 0; c < NCH; ++c) {
    if (wave == 0) __builtin_amdgcn_s_wait_tensorcnt(0);
    __syncthreads();
    if (wave == 0 && c + 1 < NCH)
      tdm_load_rows(Vb + (size_t)(c + 1) * CH * DM, (c & 1) ? lds0 : lds1, CH);
    const int bsel = c & 1;
#pragma unroll
    for (int r = 0; r < CH; ++r) {
      const int s = c * CH + r;
      float vv;
      if (s == t) vv = vr[tid];
      else        vv = sbuf[bsel][r * DM + tid];
      acc = fmaf(lw[s], vv, acc);
    }
    __syncthreads();
  }
  z_out[(size_t)bp * DM + tid] = acc;
}

// ---------------------------------------------------------------------------
// threefry2x32 (JAX PRNG, key = PRNGKey(42) = {0,42})
// ---------------------------------------------------------------------------
__device__ __forceinline__ void threefry2x32(unsigned k0, unsigned k1,
                                             unsigned x0, unsigned x1,
                                             unsigned& o0, unsigned& o1)
{
  const unsigned R[8] = {13, 15, 26, 6, 17, 29, 16, 24};
  const unsigned ks2 = k0 ^ k1 ^ 0x1BD11BDAu;
  const unsigned ks[3] = {k0, k1, ks2};
  x0 += k0; x1 += k1;
#pragma unroll
  for (int blk = 0; blk < 5; ++blk) {
#pragma unroll
    for (int r = 0; r < 4; ++r) {
      const unsigned rot = R[(blk & 1) * 4 + r];
      x0 += x1;
      x1 = (x1 << rot) | (x1 >> (32 - rot));
      x1 ^= x0;
    }
    x0 += ks[(blk + 1) % 3];
    x1 += ks[(blk + 2) % 3] + (unsigned)(blk + 1);
  }
  o0 = x0; o1 = x1;
}

// ---------------------------------------------------------------------------
// predictions -> log_w -> w = softmax(exp(log_w)) -> categorical (Gumbel) i_t
// 1 block (320 threads) per batch element b.
// ---------------------------------------------------------------------------
__global__ __launch_bounds__(320)
void weights_kernel(const float* __restrict__ r, const float* __restrict__ y,
                    const float* __restrict__ ow, const float* __restrict__ ob,
                    float* __restrict__ w_out, int* __restrict__ i_t)
{
  __shared__ float musq[PART * OUTD];
  __shared__ float wsm[PART];
  __shared__ float gum[PART * PART];
  const int b = blockIdx.x;
  const int tid = threadIdx.x;
  {
    const int p = tid >> 5, o = tid & 31;
    const float* rr = r + (size_t)(b * PART + p) * DM;
    float acc = 0.f;
    for (int d = 0; d < DM; ++d) acc = fmaf(rr[d], ow[d * OUTD + o], acc);
    acc += ob[o];
    const float mu = y[(size_t)(b * PART + p) * OUTD + o] - acc;
    musq[tid] = mu * mu;
  }
  __syncthreads();
  if (tid == 0) {
    float el[PART];
    float mxe = -1e30f;
    for (int p = 0; p < PART; ++p) {
      float s = 0.f;
      for (int o = 0; o < OUTD; ++o) s += musq[p * OUTD + o];
      // log_w = -s / (2*SIGMA_OBS) = -s ; softmax input = exp(log_w)
      el[p] = expf(-s);
      mxe = fmaxf(mxe, el[p]);
    }
    float ssum = 0.f;
    for (int p = 0; p < PART; ++p) { el[p] = expf(el[p] - mxe); ssum += el[p]; }
    for (int p = 0; p < PART; ++p) {
      wsm[p] = el[p] / ssum;
      w_out[b * PART + p] = wsm[p];
    }
  }
  __syncthreads();
  // Gumbel-max categorical over PART classes; JAX bit layout for n=3200:
  //   i < 1600  -> threefry(key,(i, i+1600)).out0
  //   i >= 1600 -> threefry(key,(i-1600, i)).out1
  if (tid < PART * PART) {
    const int p = tid / PART, c = tid - p * PART;
    const unsigned idx = (unsigned)((b * PART + p) * PART + c);
    unsigned x0, x1, o0, o1;
    bool second;
    if (idx < 1600u) { x0 = idx;         x1 = idx + 1600u; second = false; }
    else             { x0 = idx - 1600u; x1 = idx;         second = true;  }
    threefry2x32(0u, 42u, x0, x1, o0, o1);
    const unsigned bits = second ? o1 : o0;
    float u = __uint_as_float(0x3f800000u | (bits >> 9)) - 1.0f;  // [0,1)
    u = fmaxf(u, 1.17549435e-38f);
    gum[tid] = wsm[c] - logf(-logf(u));   // logits + gumbel
  }
  __syncthreads();
  if (tid < PART) {
    int best = 0;
    float bv = gum[tid * PART];
    for (int c = 1; c < PART; ++c) {
      const float g = gum[tid * PART + c];
      if (g > bv) { bv = g; best = c; }
    }
    i_t[b * PART + tid] = best;
  }
}

// ---------------------------------------------------------------------------
// Resample gather-copy of K/V/R (503 MB out): pure bandwidth, B128 + NT stores
//   s <  t : out[b,p,s] = src[b, i_t[b,p], s]
//   s == t : out[b,p,t] = rowT[b, i_t[b,p]]      (k/v/r from this step)
//   s >  t : out[b,p,s] = src[b, p, s]
// blockIdx.y selects tensor {K,V,R}; block = 4 seq rows x 64 f4 lanes.
// ---------------------------------------------------------------------------
__global__ __launch_bounds__(256)
void resample_kernel(const float* __restrict__ Kin, const float* __restrict__ Vin,
                     const float* __restrict__ Rin,
                     const float* __restrict__ kw, const float* __restrict__ vw,
                     const float* __restrict__ rw,
                     const int* __restrict__ i_t, const int* __restrict__ tptr,
                     float* __restrict__ Kout, float* __restrict__ Vout,
                     float* __restrict__ Rout)
{
  const int t     = *tptr;
  const int bid   = blockIdx.x;
  const int bp    = bid >> 7;                 // 128 chunks of 4 rows per (b,p)
  const int chunk = bid & 127;
  const int b = bp / PART, p = bp - b * PART;
  const int tid  = threadIdx.x;
  const int srow = tid >> 6;
  const int c4   = (tid & 63) << 2;
  const int s    = (chunk << 2) + srow;

  const float* src; const float* rowt; float* dst;
  if (blockIdx.y == 0)      { src = Kin; rowt = kw; dst = Kout; }
  else if (blockIdx.y == 1) { src = Vin; rowt = vw; dst = Vout; }
  else                      { src = Rin; rowt = rw; dst = Rout; }

  const int pp = (s <= t) ? i_t[bp] : p;
  f4 val;
  if (s == t) val = *(const f4*)(rowt + (size_t)(b * PART + pp) * DM + c4);
  else        val = *(const f4*)(src + ((size_t)(b * PART + pp) * SEQL + s) * DM + c4);
  __builtin_nontemporal_store(val, (f4*)(dst + ((size_t)bp * SEQL + s) * DM + c4));
}

// ---------------------------------------------------------------------------
extern "C" void kernel_launch(void* const* d_in, const int* in_sizes, int n_in,
                              void* d_out, int out_size, void* d_ws, size_t ws_size,
                              hipStream_t stream)
{
  // setup_inputs() dict order
  const float* x    = (const float*)d_in[0];
  const float* y    = (const float*)d_in[1];
  const float* Kin  = (const float*)d_in[2];
  const float* Vin  = (const float*)d_in[3];
  const float* Rin  = (const float*)d_in[4];
  const float* nq   = (const float*)d_in[5];
  const float* nk   = (const float*)d_in[6];
  const float* nv   = (const float*)d_in[7];
  const float* nz   = (const float*)d_in[8];
  const int*   tp   = (const int*)d_in[9];
  const float* wq_w = (const float*)d_in[10]; const float* wq_b = (const float*)d_in[11];
  const float* wk_w = (const float*)d_in[12]; const float* wk_b = (const float*)d_in[13];
  const float* wv_w = (const float*)d_in[14]; const float* wv_b = (const float*)d_in[15];
  const float* wz_w = (const float*)d_in[16]; const float* wz_b = (const float*)d_in[17];
  const float* ln1g = (const float*)d_in[18]; const float* ln1b = (const float*)d_in[19];
  const float* fw1  = (const float*)d_in[20]; const float* fb1  = (const float*)d_in[21];
  const float* fw2  = (const float*)d_in[22]; const float* fb2  = (const float*)d_in[23];
  const float* ln2g = (const float*)d_in[24]; const float* ln2b = (const float*)d_in[25];
  const float* ow   = (const float*)d_in[26]; const float* ob   = (const float*)d_in[27];

  // outputs, concatenated flat: r, attn_weights, K, V, R, w
  float* out      = (float*)d_out;
  float* r_out    = out;                    // 320*256
  float* attn_out = out + 81920;            // 320*512
  float* K_out    = out + 245760;           // 320*512*256
  float* V_out    = out + 42188800;
  float* R_out    = out + 84131840;
  float* w_out    = out + 126074880;        // 320

  // workspace (~2.95 MB)
  float* ws    = (float*)d_ws;
  float* q_ws  = ws;                        // 320*256
  float* k_ws  = ws + 81920;
  float* v_ws  = ws + 163840;
  float* z_ws  = ws + 245760;
  float* o_ws  = ws + 327680;               // post-LN1 "out"
  float* h_ws  = ws + 409600;               // 320*1024
  int*   it_ws = (int*)(ws + 737280);       // 320 ints

  const float sq = 0.31622776601683794f;    // sqrt(SIGMA)
  const size_t shA256  = (size_t)16 * (256 + 2) * sizeof(float);
  const size_t shLN256 = (size_t)16 * 256 * sizeof(float);

  dim3 g20(20), b256(256);
  // q/k/v projections (WMMA f32, bias + sigma*noise epilogue)
  gemm16_wmma<true><<<g20, b256, shA256, stream>>>(x, wq_w, wq_b, nq, sq, 0,
                                             nullptr, nullptr, nullptr, q_ws, 256, 256);
  gemm16_wmma<true><<<g20, b256, shA256, stream>>>(x, wk_w, wk_b, nk, sq, 0,
                                             nullptr, nullptr, nullptr, k_ws, 256, 256);
  gemm16_wmma<true><<<g20, b256, shA256, stream>>>(x, wv_w, wv_b, nv, sq, 0,
                                             nullptr, nullptr, nullptr, v_ws, 256, 256);
  // attention (TDM-staged K/V streaming; writes attn weights + z)
  attn_kernel<<<dim3(BP), b256, 0, stream>>>(Kin, Vin, q_ws, k_ws, v_ws, tp,
                                             attn_out, z_ws);
  // z-proj + noise + residual(x) + LN1
  gemm16_wmma<true><<<g20, b256, shA256 + shLN256, stream>>>(z_ws, wz_w, wz_b, nz, sq, 0,
                                                       x, ln1g, ln1b, o_ws, 256, 256);
  // FFN1 (relu)
  gemm16_wmma<true><<<g20, b256, shA256, stream>>>(o_ws, fw1, fb1, nullptr, 0.f, 1,
                                             nullptr, nullptr, nullptr, h_ws, 1024, 256);
  // FFN2 + residual(out) + LN2 -> r   (K=1024: A streamed from L2, LDS = LN stripe)
  gemm16_wmma<false><<<g20, b256, shLN256, stream>>>(h_ws, fw2, fb2, nullptr, 0.f, 0,
                                              o_ws, ln2g, ln2b, r_out, 256, 1024);
  // predictions -> log_w -> w -> i_t (threefry categorical)
  weights_kernel<<<dim3(BATCH), dim3(320), 0, stream>>>(r_out, y, ow, ob, w_out, it_ws);
  // resample gather-copy of K/V/R with row-t substitution (NT B128 stores)
  resample_kernel<<<dim3(40960, 3), b256, 0, stream>>>(Kin, Vin, Rin,
                                                       k_ws, v_ws, r_out,
                                                       it_ws, tp,
                                                       K_out, V_out, R_out);
}